// GCN_81750407512548
// MI455X (gfx1250) — compile-verified
//
#include <hip/hip_runtime.h>
#include <hip/hip_bf16.h>
#include <math.h>

#define IN_CH 256
#define HID   16

typedef float v2f __attribute__((ext_vector_type(2)));
typedef float v8f __attribute__((ext_vector_type(8)));

__device__ __forceinline__ float sigmoidf_(float v) {
    return 1.0f / (1.0f + __expf(-v));
}

// ---------------- degree (in-degree of target), then deg^-1/2 ----------------
__global__ void k_deg_count(const int* __restrict__ tgt, float* __restrict__ deg, int E) {
    int e = blockIdx.x * blockDim.x + threadIdx.x;
    if (e < E) atomicAdd(&deg[tgt[e]], 1.0f);
}

__global__ void k_deg_inv(float* __restrict__ deg, int N) {
    int i = blockIdx.x * blockDim.x + threadIdx.x;
    if (i < N) {
        float d = deg[i];
        deg[i] = (d > 0.0f) ? rsqrtf(d) : 0.0f;
    }
}

// ---------------- GEMM1: h1 = x @ W1 + b1 via V_WMMA_F32_16X16X4_F32 --------
// One wave32 computes one 16(nodes) x 16(hidden) tile, K swept 4 at a time.
// A layout: lane m=lane&15 owns row m; VGPR j holds A[m][k + 2*(lane>>4) + j]
// B layout: lane m owns col m;         VGPR j holds B[k + 2*(lane>>4) + j][m]
// D layout: VGPR i holds D[i + 8*(lane>>4)][m]
__global__ void k_gemm1(const float* __restrict__ x, const float* __restrict__ W1,
                        const float* __restrict__ b1, float* __restrict__ h1, int N) {
    const int lane = threadIdx.x & 31;
    const int wave = threadIdx.x >> 5;
    const int tile = blockIdx.x * (blockDim.x >> 5) + wave;
    const int row0 = tile * 16;
    if (row0 >= N) return;                      // wave-uniform: EXEC stays all-ones
    const int m = lane & 15;
    const int g = lane >> 4;
    int row = row0 + m; if (row >= N) row = N - 1;   // clamp loads, keep EXEC full
    const float* __restrict__ xr = x + (size_t)row * IN_CH;

    v8f acc = {};
    for (int k = 0; k < IN_CH; k += 4) {
        const int ka = k + 2 * g;
        v2f a, b;
        a.x = xr[ka];
        a.y = xr[ka + 1];
        b.x = W1[ka * HID + m];
        b.y = W1[(ka + 1) * HID + m];
        acc = __builtin_amdgcn_wmma_f32_16x16x4_f32(false, a, false, b,
                                                    (short)0, acc, false, false);
    }
    const float bias = b1[m];
    #pragma unroll
    for (int i = 0; i < 8; ++i) {
        const int r = row0 + i + 8 * g;
        if (r < N) h1[(size_t)r * HID + m] = acc[i] + bias;
    }
}

// ---------------- GEMM2: h2 = sigmoid(agg1) @ W2 + b2 (sigmoid fused) -------
__global__ void k_gemm2(const float* __restrict__ agg1, const float* __restrict__ W2,
                        const float* __restrict__ b2, float* __restrict__ h2, int N) {
    const int lane = threadIdx.x & 31;
    const int wave = threadIdx.x >> 5;
    const int tile = blockIdx.x * (blockDim.x >> 5) + wave;
    const int row0 = tile * 16;
    if (row0 >= N) return;
    const int m = lane & 15;
    const int g = lane >> 4;
    int row = row0 + m; if (row >= N) row = N - 1;
    const float* __restrict__ ar = agg1 + (size_t)row * HID;

    v8f acc = {};
    #pragma unroll
    for (int k = 0; k < HID; k += 4) {
        const int ka = k + 2 * g;
        v2f a, b;
        a.x = sigmoidf_(ar[ka]);        // each A element loaded exactly once/tile
        a.y = sigmoidf_(ar[ka + 1]);
        b.x = W2[ka * HID + m];
        b.y = W2[(ka + 1) * HID + m];
        acc = __builtin_amdgcn_wmma_f32_16x16x4_f32(false, a, false, b,
                                                    (short)0, acc, false, false);
    }
    const float bias = b2[m];
    #pragma unroll
    for (int i = 0; i < 8; ++i) {
        const int r = row0 + i + 8 * g;
        if (r < N) h2[(size_t)r * HID + m] = acc[i] + bias;
    }
}

// ---------------- edge scatter: agg[tgt] += dis[src]*dis[tgt] * h[src] ------
// lane-per-(edge,channel): 16 consecutive lanes hit 16 consecutive floats of
// one target row -> coalesced 64B atomic groups, resolved in the 192MB L2.
__global__ void k_scatter(const float* __restrict__ h, const int* __restrict__ src,
                          const int* __restrict__ tgt, const float* __restrict__ dis,
                          float* __restrict__ agg, int E) {
    long long gid = (long long)blockIdx.x * blockDim.x + threadIdx.x;
    long long e = gid >> 4;
    int c = (int)(gid & 15);
    if (e >= (long long)E) return;
    int s = src[e];
    int t = tgt[e];
    float nrm = dis[s] * dis[t];
    atomicAdd(&agg[(size_t)t * HID + c], nrm * h[(size_t)s * HID + c]);
}

// ---------------- elementwise sigmoid (final activation) --------------------
__global__ void k_sigmoid(float* __restrict__ v, long long n) {
    long long i = (long long)blockIdx.x * blockDim.x + threadIdx.x;
    if (i < n) v[i] = sigmoidf_(v[i]);
}

extern "C" void kernel_launch(void* const* d_in, const int* in_sizes, int n_in,
                              void* d_out, int out_size, void* d_ws, size_t ws_size,
                              hipStream_t stream) {
    const float* x  = (const float*)d_in[0];
    const int*   ei = (const int*)  d_in[1];
    const float* W1 = (const float*)d_in[2];
    const float* b1 = (const float*)d_in[3];
    const float* W2 = (const float*)d_in[4];
    const float* b2 = (const float*)d_in[5];

    const int N = in_sizes[0] / IN_CH;
    const int E = in_sizes[1] / 2;
    const int* src = ei;
    const int* tgt = ei + E;

    // workspace carve-out (256B aligned): deg | h1 | agg1 | h2  (~20 MB)
    char* ws = (char*)d_ws;
    size_t off = 0;
    auto carve = [&](size_t bytes) -> float* {
        float* p = (float*)(ws + off);
        off += (bytes + 255) & ~(size_t)255;
        return p;
    };
    float* deg  = carve((size_t)N * sizeof(float));
    float* h1   = carve((size_t)N * HID * sizeof(float));
    float* agg1 = carve((size_t)N * HID * sizeof(float));
    float* h2   = carve((size_t)N * HID * sizeof(float));
    float* out  = (float*)d_out;

    hipMemsetAsync(deg,  0, (size_t)N * sizeof(float), stream);
    hipMemsetAsync(agg1, 0, (size_t)N * HID * sizeof(float), stream);
    hipMemsetAsync(out,  0, (size_t)N * HID * sizeof(float), stream);

    k_deg_count<<<(E + 255) / 256, 256, 0, stream>>>(tgt, deg, E);
    k_deg_inv  <<<(N + 255) / 256, 256, 0, stream>>>(deg, N);

    const int tiles = (N + 15) / 16;
    const int wpb   = 4;                       // 4 waves / block = 128 threads
    const int gblk  = (tiles + wpb - 1) / wpb;

    k_gemm1<<<gblk, 32 * wpb, 0, stream>>>(x, W1, b1, h1, N);

    const long long ework = (long long)E * HID;
    const int sblk = (int)((ework + 255) / 256);
    k_scatter<<<sblk, 256, 0, stream>>>(h1, src, tgt, deg, agg1, E);

    k_gemm2<<<gblk, 32 * wpb, 0, stream>>>(agg1, W2, b2, h2, N);
    k_scatter<<<sblk, 256, 0, stream>>>(h2, src, tgt, deg, out, E);

    const long long nout = (long long)N * HID;
    k_sigmoid<<<(int)((nout + 255) / 256), 256, 0, stream>>>(out, nout);
}